// GAT_72301479461101
// MI455X (gfx1250) — compile-verified
//
#include <hip/hip_runtime.h>
#include <hip/hip_bf16.h>
#include <math.h>

// ---------------------------------------------------------------------------
// GAT (2x GATConv) for MI455X / gfx1250.
// Dense GEMMs use V_WMMA_F32_16X16X4_F32 (full fp32 accuracy, wave32),
// blocked 16x64 per wave with fully-unrolled K and immediate-offset loads.
// Irregular edge softmax/aggregation uses global atomics (memory-bound part).
// ---------------------------------------------------------------------------

typedef float v2f __attribute__((ext_vector_type(2)));
typedef float v8f __attribute__((ext_vector_type(8)));

// ---------------- WMMA FP32 GEMM: C[M,NCOLS] = A[M,KDIM] * B[KDIM,NCOLS] ---
// One wave computes a 16 x (16*NT) strip: NT accumulators share one A frag.
// KDIM/NCOLS are compile-time so the fully unrolled loop uses only immediate
// load offsets (no per-iteration address math).
template<int NCOLS, int KDIM, int NT>
__global__ __launch_bounds__(32)
void k_gemm_wmma_f32(const float* __restrict__ A, const float* __restrict__ B,
                     float* __restrict__ C, int M) {
    const int tileM  = blockIdx.x;
    const int tileN0 = blockIdx.y * NT;   // first of NT column tiles
    const int lane   = threadIdx.x;       // 0..31
    const int lm     = lane & 15;         // row (A) / col (B,C) within tile
    const int kg     = lane >> 4;         // K-group: 0 or 1

    int row = tileM * 16 + lm;            // A row this lane feeds
    if (row >= M) row = M - 1;            // clamp: loads stay in-bounds,
                                          // stores are guarded below
    // Per-lane base pointers (computed once; loop uses immediate offsets).
    const float* Ab = A + (size_t)row * KDIM + kg * 2;             // A[row][kg*2]
    const float* Bb = B + (size_t)(kg * 2) * NCOLS + tileN0 * 16 + lm;

    v8f acc[NT];
#pragma unroll
    for (int j = 0; j < NT; ++j) acc[j] = (v8f){};

#pragma unroll
    for (int k = 0; k < KDIM; k += 4) {
        // A 16x4 fragment: lane holds A[row][k+kg*2 .. +1] (one b64 load)
        const v2f a = *(const v2f*)(Ab + k);
#pragma unroll
        for (int j = 0; j < NT; ++j) {
            // B 4x16 fragment: lane holds B[k+kg*2][col], B[k+kg*2+1][col]
            v2f b;
            b.x = Bb[k * NCOLS + j * 16];
            b.y = Bb[k * NCOLS + NCOLS + j * 16];
            acc[j] = __builtin_amdgcn_wmma_f32_16x16x4_f32(
                /*neg_a=*/false, a, /*neg_b=*/false, b,
                /*c_mod=*/(short)0, acc[j], /*reuse_a=*/false, /*reuse_b=*/false);
        }
    }

    // C 16x16 fp32 layout: VGPR v -> row = v + kg*8, col = lm
    const int rbase = tileM * 16 + kg * 8;
#pragma unroll
    for (int j = 0; j < NT; ++j) {
        const int cbase = (tileN0 + j) * 16 + lm;
#pragma unroll
        for (int v = 0; v < 8; ++v) {
            const int r = rbase + v;
            if (r < M) C[(size_t)r * NCOLS + cbase] = acc[j][v];
        }
    }
}

// ---------------- el/er: per-(node,head) dot with attention vectors --------
__global__ void k_attn_coef(const float* __restrict__ feat,
                            const float* __restrict__ al,
                            const float* __restrict__ ar,
                            float* __restrict__ el, float* __restrict__ er,
                            int N, int H, int D) {
    const int idx = blockIdx.x * blockDim.x + threadIdx.x;
    if (idx >= N * H) return;
    const int n = idx / H, h = idx % H;
    const float* f  = feat + ((size_t)n * H + h) * D;
    const float* pl = al + h * D;
    const float* pr = ar + h * D;
    float sl = 0.0f, sr = 0.0f;
    for (int d = 0; d < D; ++d) { const float v = f[d]; sl += v * pl[d]; sr += v * pr[d]; }
    el[idx] = sl; er[idx] = sr;
}

// Monotonic float<->uint key mapping so atomicMax(u32) == float max.
__device__ __forceinline__ unsigned fkey(float f) {
    unsigned b = __float_as_uint(f);
    return b ^ (unsigned)(((int)b >> 31) | 0x80000000);
}
__device__ __forceinline__ float funkey(unsigned k) {
    unsigned b = (k & 0x80000000u) ? (k ^ 0x80000000u) : ~k;
    return __uint_as_float(b);
}

// ---------------- e = leaky_relu(el[src]+er[dst]); segment max -------------
__global__ void k_edge_max(const int* __restrict__ src, const int* __restrict__ dst,
                           const float* __restrict__ el, const float* __restrict__ er,
                           float* __restrict__ ebuf, unsigned* __restrict__ emax,
                           int E, int H) {
    const int idx = blockIdx.x * blockDim.x + threadIdx.x;
    if (idx >= E * H) return;
    const int e = idx / H, h = idx % H;
    const int s = src[e], d = dst[e];
    float v = el[s * H + h] + er[d * H + h];
    v = (v > 0.0f) ? v : 0.2f * v;               // leaky_relu, slope 0.2
    ebuf[idx] = v;
    atomicMax(&emax[d * H + h], fkey(v));
}

// ---------------- ex = exp(e - max); segment sum of ex ---------------------
__global__ void k_edge_exp(const int* __restrict__ dst,
                           float* __restrict__ ebuf, const unsigned* __restrict__ emax,
                           float* __restrict__ denom, int E, int H) {
    const int idx = blockIdx.x * blockDim.x + threadIdx.x;
    if (idx >= E * H) return;
    const int e = idx / H, h = idx % H;
    const int d = dst[e];
    const float ex = expf(ebuf[idx] - funkey(emax[d * H + h]));
    ebuf[idx] = ex;
    atomicAdd(&denom[d * H + h], ex);
}

// ---------------- out[dst] += feat[src] * alpha (one block per edge) -------
__global__ void k_aggregate(const int* __restrict__ src, const int* __restrict__ dst,
                            const float* __restrict__ feat, const float* __restrict__ ebuf,
                            const float* __restrict__ denom, float* __restrict__ out,
                            int H, int D) {
    const int e = blockIdx.x;
    const int t = threadIdx.x;              // 0 .. H*D-1
    const int s = src[e], d = dst[e];
    __shared__ float salpha[8];
    if (t < H) salpha[t] = ebuf[e * H + t] / denom[d * H + t];
    __syncthreads();
    const int h = t / D;
    atomicAdd(&out[(size_t)d * H * D + t], feat[(size_t)s * H * D + t] * salpha[h]);
}

// ---------------- ELU in place ---------------------------------------------
__global__ void k_elu(float* __restrict__ x, int n) {
    const int idx = blockIdx.x * blockDim.x + threadIdx.x;
    if (idx >= n) return;
    const float v = x[idx];
    x[idx] = (v > 0.0f) ? v : (expf(v) - 1.0f);
}

// ---------------- final: logits = agg1 + residual --------------------------
__global__ void k_add(const float* __restrict__ a, const float* __restrict__ b,
                      float* __restrict__ o, int n) {
    const int idx = blockIdx.x * blockDim.x + threadIdx.x;
    if (idx < n) o[idx] = a[idx] + b[idx];
}

extern "C" void kernel_launch(void* const* d_in, const int* in_sizes, int n_in,
                              void* d_out, int out_size, void* d_ws, size_t ws_size,
                              hipStream_t stream) {
    const float* x    = (const float*)d_in[0];
    const int*   src  = (const int*)  d_in[1];
    const int*   dst  = (const int*)  d_in[2];
    const float* W0   = (const float*)d_in[3];
    const float* al0  = (const float*)d_in[4];
    const float* ar0  = (const float*)d_in[5];
    const float* W1   = (const float*)d_in[6];
    const float* al1  = (const float*)d_in[7];
    const float* ar1  = (const float*)d_in[8];
    const float* Wres = (const float*)d_in[9];
    float* out = (float*)d_out;

    const int N = in_sizes[0] / 256;   // 50000 nodes
    const int E = in_sizes[1];         // 850000 edges (incl. self loops)
    const int F = 256;                 // input feature dim / hidden dim

    // ---- workspace carve-out (all fp32 unless noted), 256B aligned --------
    char* ws = (char*)d_ws;
    size_t off = 0;
    auto alloc = [&](size_t elems) -> float* {
        float* p = (float*)(ws + off);
        off += ((elems * sizeof(float) + 255) / 256) * 256;
        return p;
    };
    float*    feat0  = alloc((size_t)N * F);      // x @ W0      [N,4,64]
    float*    hbuf   = alloc((size_t)N * F);      // layer0 out / layer1 in
    float*    el0    = alloc((size_t)N * 4);
    float*    er0    = alloc((size_t)N * 4);
    unsigned* emax0  = (unsigned*)alloc((size_t)N * 4);
    float*    denom0 = alloc((size_t)N * 4);
    float*    ebuf0  = alloc((size_t)E * 4);      // e, then exp(e - max)
    float*    feat1  = alloc((size_t)N * 32);     // h @ W1
    float*    resb   = alloc((size_t)N * 32);     // h @ Wres
    float*    el1    = alloc((size_t)N);
    float*    er1    = alloc((size_t)N);
    unsigned* emax1  = (unsigned*)alloc((size_t)N);
    float*    denom1 = alloc((size_t)N);
    float*    ebuf1  = alloc((size_t)E);
    float*    agg1   = alloc((size_t)N * 32);

    // ---- zero accumulators (ws is NOT re-poisoned between replays) --------
    hipMemsetAsync(hbuf,   0, (size_t)N * F  * sizeof(float), stream);
    hipMemsetAsync(emax0,  0, (size_t)N * 4  * sizeof(unsigned), stream);
    hipMemsetAsync(denom0, 0, (size_t)N * 4  * sizeof(float), stream);
    hipMemsetAsync(emax1,  0, (size_t)N      * sizeof(unsigned), stream);
    hipMemsetAsync(denom1, 0, (size_t)N      * sizeof(float), stream);
    hipMemsetAsync(agg1,   0, (size_t)N * 32 * sizeof(float), stream);

    const int mt = (N + 15) / 16;

    // ================= Layer 0: 4 heads x 64 dims ==========================
    // feat0 = x @ W0   (WMMA fp32, 16x64 strip per wave)
    k_gemm_wmma_f32<256, 256, 4><<<dim3(mt, 4), 32, 0, stream>>>(x, W0, feat0, N);
    // el/er
    k_attn_coef<<<(N * 4 + 255) / 256, 256, 0, stream>>>(feat0, al0, ar0, el0, er0, N, 4, 64);
    // edge scores + segment max
    k_edge_max<<<((size_t)E * 4 + 255) / 256, 256, 0, stream>>>(src, dst, el0, er0, ebuf0, emax0, E, 4);
    // exp + segment sum
    k_edge_exp<<<((size_t)E * 4 + 255) / 256, 256, 0, stream>>>(dst, ebuf0, emax0, denom0, E, 4);
    // weighted aggregation into hbuf
    k_aggregate<<<E, 256, 0, stream>>>(src, dst, feat0, ebuf0, denom0, hbuf, 4, 64);
    // ELU activation (layer0 output -> layer1 input, flattened [N,256])
    k_elu<<<((size_t)N * F + 255) / 256, 256, 0, stream>>>(hbuf, N * F);

    // ================= Layer 1: 1 head x 32 dims ===========================
    // feat1 = h @ W1 ; resb = h @ Wres   (WMMA fp32, 16x32 strip per wave)
    k_gemm_wmma_f32<32, 256, 2><<<dim3(mt, 1), 32, 0, stream>>>(hbuf, W1,   feat1, N);
    k_gemm_wmma_f32<32, 256, 2><<<dim3(mt, 1), 32, 0, stream>>>(hbuf, Wres, resb,  N);
    // el/er
    k_attn_coef<<<(N + 255) / 256, 256, 0, stream>>>(feat1, al1, ar1, el1, er1, N, 1, 32);
    // edge scores + segment max
    k_edge_max<<<(E + 255) / 256, 256, 0, stream>>>(src, dst, el1, er1, ebuf1, emax1, E, 1);
    // exp + segment sum
    k_edge_exp<<<(E + 255) / 256, 256, 0, stream>>>(dst, ebuf1, emax1, denom1, E, 1);
    // weighted aggregation (one wave per edge)
    k_aggregate<<<E, 32, 0, stream>>>(src, dst, feat1, ebuf1, denom1, agg1, 1, 32);
    // logits = agg1 + residual   (mean over 1 head is identity)
    k_add<<<((size_t)N * 32 + 255) / 256, 256, 0, stream>>>(agg1, resb, out, N * 32);
}